// RecurrentGCN_86088324481398
// MI455X (gfx1250) — compile-verified
//
#include <hip/hip_runtime.h>

// DCRNN cell on MI455X (gfx1250). H0==0 => R/Wr dead; h=(1-sigmoid(z))*tanh(c).
// Pipeline: zero ws -> edge degree atomics -> invert -> edge scatter (S_o,S_i)
// -> build 36x128 combined weight -> fused WMMA GEMM + gate epilogue + 64->4 proj.
// GEMM uses V_WMMA_F32_16X16X4_F32 (fp32-exact; compute is ~1.5 GFLOP, memory-bound).

typedef __attribute__((ext_vector_type(2))) float v2f;
typedef __attribute__((ext_vector_type(8))) float v8f;

#define N_NODES 10000
#define BATCH   16
#define FEAT    12
#define NEDGE   80000
#define HID     64
#define KDIM    36     // 12 (x) + 12 (To) + 12 (Ti)
#define NCOLS   128    // 64 z-gate + 64 c-gate
#define FIN     76     // F + HIDDEN (reference weight leading dim)

__global__ void k_zero(float* __restrict__ p, int n) {
    int i = blockIdx.x * blockDim.x + threadIdx.x;
    if (i < n) p[i] = 0.0f;
}

__global__ void k_deg(const int* __restrict__ row, const int* __restrict__ col,
                      const float* __restrict__ ew,
                      float* __restrict__ dego, float* __restrict__ degi) {
    int e = blockIdx.x * blockDim.x + threadIdx.x;
    if (e < NEDGE) {
        float w = ew[e];
        atomicAdd(dego + row[e], w);
        atomicAdd(degi + col[e], w);
    }
}

__global__ void k_inv(float* __restrict__ dego, float* __restrict__ degi) {
    int i = blockIdx.x * blockDim.x + threadIdx.x;
    if (i < N_NODES) {
        float a = dego[i]; dego[i] = (a > 0.0f) ? 1.0f / a : 0.0f;
        float b = degi[i]; degi[i] = (b > 0.0f) ? 1.0f / b : 0.0f;
    }
}

// S_o[b,row,:] += ew * x[b,col,:] ; S_i[b,col,:] += ew * x[b,row,:]
__global__ void k_scatter(const float* __restrict__ x,
                          const int* __restrict__ row, const int* __restrict__ col,
                          const float* __restrict__ ew,
                          float* __restrict__ so, float* __restrict__ si) {
    long t = (long)blockIdx.x * blockDim.x + threadIdx.x;
    if (t >= (long)NEDGE * BATCH) return;
    int e = (int)(t % NEDGE);
    int b = (int)(t / NEDGE);
    int ri = row[e], ci = col[e];
    float w = ew[e];
    const float* xc = x + ((long)b * N_NODES + ci) * FEAT;
    const float* xr = x + ((long)b * N_NODES + ri) * FEAT;
    float* sod = so + ((long)b * N_NODES + ri) * FEAT;
    float* sid = si + ((long)b * N_NODES + ci) * FEAT;
#pragma unroll
    for (int f = 0; f < FEAT; ++f) {
        atomicAdd(sod + f, w * xc[f]);
        atomicAdd(sid + f, w * xr[f]);
    }
}

// wcat[k][j]: k 0..11 -> W[0,0]+W[1,0]; 12..23 -> W[0,1]; 24..35 -> W[1,1].
// j 0..63 from Wz, 64..127 from Wc. W flattened [2,2,FIN,HID].
__global__ void k_wcat(const float* __restrict__ Wz, const float* __restrict__ Wc,
                       float* __restrict__ wcat) {
    int t = blockIdx.x * blockDim.x + threadIdx.x;
    if (t >= KDIM * NCOLS) return;
    int k = t / NCOLS, j = t % NCOLS;
    const float* W = (j < HID) ? Wz : Wc;
    int jj = j & (HID - 1);
    float v;
    if (k < 12)      v = W[(0 * FIN + k) * HID + jj] + W[(2 * FIN + k) * HID + jj];
    else if (k < 24) v = W[(1 * FIN + (k - 12)) * HID + jj];
    else             v = W[(3 * FIN + (k - 24)) * HID + jj];
    wcat[t] = v;
}

// One wave per 16-row tile; 8 waves per block; 1250 blocks cover 160000 rows exactly.
__launch_bounds__(256)
__global__ void k_main(const float* __restrict__ x,
                       const float* __restrict__ so, const float* __restrict__ si,
                       const float* __restrict__ invo, const float* __restrict__ invi,
                       const float* __restrict__ wcat,
                       const float* __restrict__ bz, const float* __restrict__ bc,
                       const float* __restrict__ wlin, const float* __restrict__ blin,
                       float* __restrict__ out) {
    __shared__ float lw[KDIM * NCOLS];   // 36x128 weights
    __shared__ float lb[2 * HID];        // bz | bc
    __shared__ float lwl[HID * 4];       // W_lin
    __shared__ float lbl[4];             // b_lin
    __shared__ float lh[8][16][65];      // per-wave h, padded vs bank conflicts

    int tid = threadIdx.x;
    for (int i = tid; i < KDIM * NCOLS; i += 256) lw[i] = wcat[i];
    if (tid < HID)               lb[tid] = bz[tid];
    else if (tid < 2 * HID)      lb[tid] = bc[tid - HID];
    lwl[tid] = wlin[tid];        // exactly 256 elements
    if (tid < 4) lbl[tid] = blin[tid];
    __syncthreads();

    int wave = tid >> 5, lane = tid & 31;
    int m = lane & 15, hi = lane >> 4;
    int tile = blockIdx.x * 8 + wave;
    int row0 = tile * 16;
    int r = row0 + m;            // this lane's A-matrix row (b*N + n)
    int n = r % N_NODES;
    float io = invo[n], ii = invi[n];

    const float* xr  = x  + (long)r * FEAT;
    const float* sor = so + (long)r * FEAT;
    const float* sir = si + (long)r * FEAT;
    int ko = 2 * hi;             // lanes 16-31 hold K+2,K+3 (32-bit A layout)

    v2f a[9];
#pragma unroll
    for (int s = 0; s < 3; ++s) {
        a[s]     = *(const v2f*)(xr  + 4 * s + ko);
        v2f t1   = *(const v2f*)(sor + 4 * s + ko);
        a[3 + s] = t1 * io;      // fold 1/deg_out into A
        v2f t2   = *(const v2f*)(sir + 4 * s + ko);
        a[6 + s] = t2 * ii;      // fold 1/deg_in into A
    }

    v8f accs[8];
#pragma unroll
    for (int ct = 0; ct < 8; ++ct) {
        v8f acc = {};
#pragma unroll
        for (int s = 0; s < 9; ++s) {
            int kb = 4 * s + ko;
            v2f bf;
            bf[0] = lw[kb * NCOLS + ct * 16 + m];
            bf[1] = lw[(kb + 1) * NCOLS + ct * 16 + m];
            acc = __builtin_amdgcn_wmma_f32_16x16x4_f32(
                false, a[s], false, bf, (short)0, acc, false, false);
        }
        accs[ct] = acc;
    }

    // Gate fusion: col j (z) pairs with col j+64 (c); C/D layout: M = v + 8*hi, N = m.
    float (*hrow)[65] = lh[wave];
#pragma unroll
    for (int ct = 0; ct < 4; ++ct) {
        int colz = ct * 16 + m;
#pragma unroll
        for (int v = 0; v < 8; ++v) {
            float zl = accs[ct][v]     + lb[colz];
            float cl = accs[ct + 4][v] + lb[HID + colz];
            float zs = 1.0f / (1.0f + __expf(-zl));
            float h  = (1.0f - zs) * tanhf(cl);
            hrow[v + 8 * hi][colz] = fmaxf(h, 0.0f);
        }
    }
    __syncthreads();

    // Final 64 -> 4 projection: lane covers (row=m, p in {2*hi, 2*hi+1}).
    int p0 = 2 * hi;
    float acc0 = lbl[p0], acc1 = lbl[p0 + 1];
#pragma unroll 8
    for (int j = 0; j < HID; ++j) {
        float hv = hrow[m][j];
        acc0 += hv * lwl[j * 4 + p0];
        acc1 += hv * lwl[j * 4 + p0 + 1];
    }
    float* orow = out + (long)(row0 + m) * 4;
    orow[p0]     = acc0;
    orow[p0 + 1] = acc1;
}

extern "C" void kernel_launch(void* const* d_in, const int* in_sizes, int n_in,
                              void* d_out, int out_size, void* d_ws, size_t ws_size,
                              hipStream_t stream) {
    const float* x  = (const float*)d_in[0];
    const int*   ei = (const int*)d_in[1];      // jax default x64 off -> int32
    const float* ew = (const float*)d_in[2];
    const float* Wz = (const float*)d_in[3];
    const float* bz = (const float*)d_in[4];
    // d_in[5]=Wr, d_in[6]=br: dead (H0 == 0 makes R irrelevant)
    const float* Wc = (const float*)d_in[7];
    const float* bc = (const float*)d_in[8];
    const float* Wl = (const float*)d_in[9];
    const float* bl = (const float*)d_in[10];
    float* out = (float*)d_out;

    const int* row = ei;
    const int* col = ei + NEDGE;

    float* ws   = (float*)d_ws;
    float* dego = ws;                                   // N (-> inv_out in place)
    float* degi = dego + N_NODES;                       // N (-> inv_in in place)
    float* so   = degi + N_NODES;                       // B*N*F
    float* si   = so + (size_t)BATCH * N_NODES * FEAT;  // B*N*F
    float* wcat = si + (size_t)BATCH * N_NODES * FEAT;  // 36*128
    int zero_count = 2 * N_NODES + 2 * BATCH * N_NODES * FEAT;

    k_zero<<<(zero_count + 255) / 256, 256, 0, stream>>>(ws, zero_count);
    k_deg<<<(NEDGE + 255) / 256, 256, 0, stream>>>(row, col, ew, dego, degi);
    k_inv<<<(N_NODES + 255) / 256, 256, 0, stream>>>(dego, degi);
    {
        long nt = (long)NEDGE * BATCH;
        k_scatter<<<(int)((nt + 255) / 256), 256, 0, stream>>>(x, row, col, ew, so, si);
    }
    k_wcat<<<(KDIM * NCOLS + 255) / 256, 256, 0, stream>>>(Wz, Wc, wcat);
    k_main<<<(BATCH * N_NODES) / 128, 256, 0, stream>>>(
        x, so, si, dego, degi, wcat, bz, bc, Wl, bl, out);
}